// CenterLoss_7232724926888
// MI455X (gfx1250) — compile-verified
//
#include <hip/hip_runtime.h>

typedef float v2f __attribute__((ext_vector_type(2)));
typedef float v8f __attribute__((ext_vector_type(8)));

// Per-row squared distance ||x_i - centers[labels[i]]||^2 via
// V_WMMA_F32_16X16X4_F32 on difference vectors (diagonal of D accumulates
// sum(diff^2)). Block = 128 threads = 4 waves; block owns 16 rows; wave w
// handles K-chunk [w*D/4, (w+1)*D/4).
template <int D_CT>
__global__ __launch_bounds__(128) void center_dist_wmma_kernel(
    const float* __restrict__ x,
    const long long* __restrict__ labels,
    const float* __restrict__ centers,
    float* __restrict__ dist,
    int D_rt)
{
  const int D = (D_CT > 0) ? D_CT : D_rt;  // compile-time when D_CT > 0

  const int tile  = blockIdx.x;            // 16 rows per tile
  const int lane  = threadIdx.x & 31;
  const int wave  = threadIdx.x >> 5;      // 0..3
  const int m     = lane & 15;             // row (A) == col (B) for this lane
  const int khalf = (lane >> 4) * 2;       // lanes 0-15: k0,k0+1 ; lanes 16-31: k0+2,k0+3

  const int row = tile * 16 + m;
  const long long lbl = labels[row];
  const float* __restrict__ xrow = x + (size_t)row * (size_t)D;
  const float* __restrict__ crow = centers + (size_t)lbl * (size_t)D;

  const int dquarter = D >> 2;             // per-wave K chunk (128 for D=512)
  const int kbase = wave * dquarter + khalf;

  v8f acc = {};
  #pragma unroll 8
  for (int i = 0; i < dquarter; i += 4) {
    const int k = kbase + i;
    v2f xa = *(const v2f*)(xrow + k);
    v2f ca = *(const v2f*)(crow + k);
    v2f d;
    d.x = xa.x - ca.x;
    d.y = xa.y - ca.y;
    // A = diff (16xK), B = diff^T (Kx16): identical lane layout, so the same
    // v2f serves as both operands; diag(D) accumulates sum(diff^2).
    acc = __builtin_amdgcn_wmma_f32_16x16x4_f32(
        /*neg_a=*/false, d, /*neg_b=*/false, d,
        /*c_mod=*/(short)0, acc, /*reuse_a=*/false, /*reuse_b=*/false);
  }

  // Diagonal extraction from 16x16 f32 C/D layout:
  //   row r (0..7)  -> lane r,      VGPR r
  //   row r (8..15) -> lane r + 16, VGPR r - 8
  __shared__ float lds_d[4 * 16];
  float dg = 0.0f;
  int r = -1;
  if (lane < 8) {
    r = lane;
    dg = acc[lane];
  } else if (lane >= 24) {
    r = lane - 16;
    dg = acc[lane - 24];
  }
  if (r >= 0) lds_d[wave * 16 + r] = dg;
  __syncthreads();

  if (threadIdx.x < 16) {
    const int t = threadIdx.x;
    float s = lds_d[t] + lds_d[16 + t] + lds_d[32 + t] + lds_d[48 + t];
    dist[tile * 16 + t] = s;
  }
}

// Deterministic clip + mean over N per-row distances (single block tree).
__global__ __launch_bounds__(256) void clip_mean_kernel(
    const float* __restrict__ dist, float* __restrict__ out, int N)
{
  __shared__ float sm[256];
  float s = 0.0f;
  for (int i = threadIdx.x; i < N; i += 256) {
    float d = dist[i];
    d = fminf(fmaxf(d, 1e-12f), 1e12f);
    s += d;
  }
  sm[threadIdx.x] = s;
  __syncthreads();
  for (int off = 128; off > 0; off >>= 1) {
    if ((int)threadIdx.x < off) sm[threadIdx.x] += sm[threadIdx.x + off];
    __syncthreads();
  }
  if (threadIdx.x == 0) out[0] = sm[0] / (float)N;
}

extern "C" void kernel_launch(void* const* d_in, const int* in_sizes, int n_in,
                              void* d_out, int out_size, void* d_ws, size_t ws_size,
                              hipStream_t stream) {
  const float*     x       = (const float*)d_in[0];      // (N, D) fp32
  const long long* labels  = (const long long*)d_in[1];  // (N,)   int64
  const float*     centers = (const float*)d_in[2];      // (C, D) fp32

  const int N = in_sizes[1];             // 4096
  const int D = in_sizes[0] / N;         // 512

  float* ws  = (float*)d_ws;             // N floats of per-row distances
  float* out = (float*)d_out;            // scalar fp32

  const int tiles = N / 16;              // 256 blocks, 16 rows each
  if (D == 512) {
    center_dist_wmma_kernel<512><<<tiles, 128, 0, stream>>>(x, labels, centers, ws, D);
  } else {
    center_dist_wmma_kernel<0><<<tiles, 128, 0, stream>>>(x, labels, centers, ws, D);
  }
  clip_mean_kernel<<<1, 256, 0, stream>>>(ws, out, N);
}